// GaussianModel_79886391706255
// MI455X (gfx1250) — compile-verified
//
#include <hip/hip_runtime.h>

// Streaming, memory-bound Gaussian-preprocess kernel for MI455X (gfx1250).
//
// Roofline: reads 32 B/point contiguous + ~192 B/point of cacheline traffic
// for the strided features[:,0] gather, writes 68 B/point -> ~580 MB total
// vs ~0.2 GFLOP => DRAM-bound by >20x (floor ~25 us at 23.3 TB/s).
// Therefore: no WMMA (per-point 3x3 products are block-diagonal across 2M
// points; <4% tile utilization), no TDM/LDS staging (zero data reuse).
// The CDNA5 levers that matter are cache policy (TH=NT on all single-touch
// streams so the 192 MB L2 isn't thrashed) and global_prefetch_b8 to cover
// the latency of the strided features gather.

typedef float v4f __attribute__((ext_vector_type(4)));

__global__ __launch_bounds__(256) void gs_preprocess_kernel(
    const float* __restrict__ scaling_raw,   // (N,3)
    const float* __restrict__ rotation,      // (N,4)
    const float* __restrict__ opacity_raw,   // (N,1)
    const float* __restrict__ features,      // (N,16,3) -> row stride 48 floats
    float* __restrict__ out_cov3d,           // (N,3,3)
    float* __restrict__ out_cov2d,           // (N,2,2)
    float* __restrict__ out_color,           // (N,3)
    float* __restrict__ out_opacity,         // (N,1)
    int N)
{
    const int tid    = blockIdx.x * blockDim.x + threadIdx.x;
    const int stride = gridDim.x * blockDim.x;

    // Hoisted per-thread stream pointers: loop advances them with pure adds
    // (no per-iteration 64-bit mads).
    const v4f*   rot_p  = reinterpret_cast<const v4f*>(rotation) + tid;
    const float* scl_p  = scaling_raw + (size_t)tid * 3;
    const float* opa_p  = opacity_raw + tid;
    const float* fea_p  = features    + (size_t)tid * 48;
    float*       c3_p   = out_cov3d   + (size_t)tid * 9;
    float*       c2_p   = out_cov2d   + (size_t)tid * 4;
    float*       col_p  = out_color   + (size_t)tid * 3;
    float*       opo_p  = out_opacity + tid;

    for (int n = tid; n < N; n += stride) {
        // Prefetch the next iteration's strided features row (the one access
        // the coalescer can't densify). Clamped index: no branch, and stays
        // in-bounds (emits global_prefetch_b8, gfx1250 path).
        int np = n + stride;
        np = (np < N) ? np : n;
        __builtin_prefetch(features + (size_t)np * 48, 0, 0);

        // ---- quaternion -> rotation matrix (normalized) ----
        const v4f q = __builtin_nontemporal_load(rot_p);   // 16B-aligned b128
        const float inv = rsqrtf(q.x*q.x + q.y*q.y + q.z*q.z + q.w*q.w);
        const float w = q.x * inv, x = q.y * inv, y = q.z * inv, z = q.w * inv;

        const float xx = x*x, yy = y*y, zz = z*z;
        const float xy = x*y, xz = x*z, yz = y*z;
        const float wx = w*x, wy = w*y, wz = w*z;

        const float R00 = 1.0f - 2.0f*(yy + zz);
        const float R01 = 2.0f*(xy - wz);
        const float R02 = 2.0f*(xz + wy);
        const float R10 = 2.0f*(xy + wz);
        const float R11 = 1.0f - 2.0f*(xx + zz);
        const float R12 = 2.0f*(yz - wx);
        const float R20 = 2.0f*(xz - wy);
        const float R21 = 2.0f*(yz + wx);
        const float R22 = 1.0f - 2.0f*(xx + yy);

        // ---- scales: sigma = exp(s), d = sigma^2 ----
        const float sg0 = expf(__builtin_nontemporal_load(scl_p + 0));
        const float sg1 = expf(__builtin_nontemporal_load(scl_p + 1));
        const float sg2 = expf(__builtin_nontemporal_load(scl_p + 2));
        const float d0 = sg0*sg0, d1 = sg1*sg1, d2 = sg2*sg2;

        // ---- cov3d = R * diag(d) * R^T (symmetric; 6 unique terms) ----
        const float c00 = R00*R00*d0 + R01*R01*d1 + R02*R02*d2;
        const float c01 = R00*R10*d0 + R01*R11*d1 + R02*R12*d2;
        const float c02 = R00*R20*d0 + R01*R21*d1 + R02*R22*d2;
        const float c11 = R10*R10*d0 + R11*R11*d1 + R12*R12*d2;
        const float c12 = R10*R20*d0 + R11*R21*d1 + R12*R22*d2;
        const float c22 = R20*R20*d0 + R21*R21*d1 + R22*R22*d2;

        const float var = (c00 + c11 + c22) * (1.0f / 3.0f);

        // ---- streaming stores (non-temporal: single-touch output) ----
        __builtin_nontemporal_store(c00, c3_p + 0);
        __builtin_nontemporal_store(c01, c3_p + 1);
        __builtin_nontemporal_store(c02, c3_p + 2);
        __builtin_nontemporal_store(c01, c3_p + 3);
        __builtin_nontemporal_store(c11, c3_p + 4);
        __builtin_nontemporal_store(c12, c3_p + 5);
        __builtin_nontemporal_store(c02, c3_p + 6);
        __builtin_nontemporal_store(c12, c3_p + 7);
        __builtin_nontemporal_store(c22, c3_p + 8);

        __builtin_nontemporal_store(var,  c2_p + 0);
        __builtin_nontemporal_store(0.0f, c2_p + 1);
        __builtin_nontemporal_store(0.0f, c2_p + 2);
        __builtin_nontemporal_store(var,  c2_p + 3);

        // ---- color = features[n, 0, :] ----
        // Row offset 192*n bytes -> 16B aligned; the 4th lane of the b128
        // read touches features[n,1,0] at worst, always in-bounds.
        const v4f f = __builtin_nontemporal_load(
            reinterpret_cast<const v4f*>(fea_p));
        __builtin_nontemporal_store(f.x, col_p + 0);
        __builtin_nontemporal_store(f.y, col_p + 1);
        __builtin_nontemporal_store(f.z, col_p + 2);

        // ---- opacity = sigmoid(opacity_raw) ----
        const float o = __builtin_nontemporal_load(opa_p);
        __builtin_nontemporal_store(1.0f / (1.0f + expf(-o)), opo_p);

        // ---- advance all streams by one grid-stride (adds only) ----
        rot_p += stride;
        scl_p += (size_t)stride * 3;
        opa_p += stride;
        fea_p += (size_t)stride * 48;
        c3_p  += (size_t)stride * 9;
        c2_p  += (size_t)stride * 4;
        col_p += (size_t)stride * 3;
        opo_p += stride;
    }
}

extern "C" void kernel_launch(void* const* d_in, const int* in_sizes, int n_in,
                              void* d_out, int out_size, void* d_ws, size_t ws_size,
                              hipStream_t stream) {
    // setup_inputs order:
    // 0: xyz (N,3)        -- unused by reference outputs
    // 1: scaling_raw (N,3)
    // 2: rotation (N,4)
    // 3: opacity_raw (N,1)
    // 4: features (N,16,3)
    // 5: viewmatrix (4,4) -- unused
    // 6: projmatrix (4,4) -- unused
    const float* scaling  = (const float*)d_in[1];
    const float* rotation = (const float*)d_in[2];
    const float* opacity  = (const float*)d_in[3];
    const float* features = (const float*)d_in[4];

    const int N = in_sizes[3];  // opacity_raw has exactly N elements

    float* out        = (float*)d_out;
    float* out_cov3d  = out;                      // 9N
    float* out_cov2d  = out + (size_t)9  * N;     // 4N
    float* out_color  = out + (size_t)13 * N;     // 3N
    float* out_opac   = out + (size_t)16 * N;     // 1N

    const int threads = 256;                       // 8 wave32 waves / block
    int blocks = (N + threads - 1) / threads;
    if (blocks > 4096) blocks = 4096;              // persistent grid: the
                                                   // stride-ahead prefetch
                                                   // targets real future work
    gs_preprocess_kernel<<<blocks, threads, 0, stream>>>(
        scaling, rotation, opacity, features,
        out_cov3d, out_cov2d, out_color, out_opac, N);
}